// NoOtherStartingStatesLoss_22179211117290
// MI455X (gfx1250) — compile-verified
//
#include <hip/hip_runtime.h>
#include <hip/hip_bf16.h>
#include <stdint.h>

// Problem constants (fixed by the reference).
#define N_TRANS   1024
#define N_STATES  512
#define N_ROWS    (N_TRANS * N_STATES)   // 524288 rows of 512 f32
#define WAVESZ    32
#define WPB       8                      // waves per block
#define BLOCK_T   (WAVESZ * WPB)         // 256 threads
#define GRID_B    4096                   // 32768 waves -> 16 rows/wave

typedef int v4i __attribute__((ext_vector_type(4)));
typedef __attribute__((address_space(1))) v4i gas_v4i;
typedef __attribute__((address_space(3))) v4i las_v4i;

#if defined(__has_builtin)
#if __has_builtin(__builtin_amdgcn_global_load_async_to_lds_b128)
#define HAVE_ASYNC_LDS 1
#endif
#endif
#ifndef HAVE_ASYNC_LDS
#define HAVE_ASYNC_LDS 0
#endif

// Copy 16 bytes per lane from global to LDS.
__device__ __forceinline__ void async_copy_b128(const float* g, float* l) {
#if HAVE_ASYNC_LDS
  __builtin_amdgcn_global_load_async_to_lds_b128(
      (gas_v4i*)(uintptr_t)g,
      (las_v4i*)(unsigned)(uintptr_t)l,
      /*imm offset*/0, /*cpol*/0);
#else
  // Synchronous fallback: LDS ops are in-order within a wave, so later
  // ds_reads observe these writes without an explicit barrier.
  *(float4*)l = *(const float4*)g;
#endif
}

#if HAVE_ASYNC_LDS
#define WAIT_ASYNC(n) asm volatile("s_wait_asynccnt %0" :: "n"(n) : "memory")
#else
#define WAIT_ASYNC(n) do {} while (0)
#endif

// Stage one 512-float row into LDS: 4 async instructions x 512 contiguous bytes.
__device__ __forceinline__ void stage_row(const float* __restrict__ g,
                                          float* __restrict__ l, int lane) {
#pragma unroll
  for (int k = 0; k < 4; ++k)
    async_copy_b128(g + k * 128 + lane * 4, l + k * 128 + lane * 4);
}

__device__ __forceinline__ float wave_max(float v) {
#pragma unroll
  for (int m = 16; m >= 1; m >>= 1) v = fmaxf(v, __shfl_xor(v, m, WAVESZ));
  return v;
}
__device__ __forceinline__ float wave_sum(float v) {
#pragma unroll
  for (int m = 16; m >= 1; m >>= 1) v += __shfl_xor(v, m, WAVESZ);
  return v;
}

__global__ void __launch_bounds__(BLOCK_T)
no_other_start_loss_kernel(const float* __restrict__ T,
                           const int* __restrict__ sel,
                           float* __restrict__ out) {
  __shared__ __align__(16) float stage[WPB * 2 * N_STATES];  // 32 KB
  __shared__ float block_acc;
  if (threadIdx.x == 0) block_acc = 0.0f;
  __syncthreads();

  const int lane  = threadIdx.x & (WAVESZ - 1);
  const int wid   = threadIdx.x >> 5;
  const int gwave = blockIdx.x * WPB + wid;
  const int nwave = gridDim.x * WPB;

  float* const buf0 = &stage[wid * 2 * N_STATES];
  float* const buf1 = buf0 + N_STATES;

  // Selected columns: lane j (j<4) owns sel[j].
  const bool pick  = lane < 4;
  const int  mycol = sel[pick ? lane : 0];

  float acc = 0.0f;

  int row = gwave;
  int cur = 0;
  if (row < N_ROWS) stage_row(T + (size_t)row * N_STATES, buf0, lane);

  while (row < N_ROWS) {
    const int next = row + nwave;
    float* bufc = cur ? buf1 : buf0;
    float* bufn = cur ? buf0 : buf1;

    if (next < N_ROWS) {
      stage_row(T + (size_t)next * N_STATES, bufn, lane);
      WAIT_ASYNC(4);   // current row's 4 transfers done; prefetch in flight
    } else {
      WAIT_ASYNC(0);
    }

    // ---- row softmax statistics (16 elements per lane) ----
    float x[16];
    const float4* b4 = (const float4*)bufc;
#pragma unroll
    for (int k = 0; k < 4; ++k) {
      float4 v = b4[k * 32 + lane];
      x[4 * k + 0] = v.x; x[4 * k + 1] = v.y;
      x[4 * k + 2] = v.z; x[4 * k + 3] = v.w;
    }
    float mx = x[0];
#pragma unroll
    for (int i = 1; i < 16; ++i) mx = fmaxf(mx, x[i]);
    mx = wave_max(mx);

    float s = 0.0f;
#pragma unroll
    for (int i = 0; i < 16; ++i) s += __expf(x[i] - mx);
    s = wave_sum(s);
    const float invd = __builtin_amdgcn_rcpf(s);

    // ---- selected columns (exclude diagonal s1 == c) ----
    const int s1 = row & (N_STATES - 1);
    if (pick) {
      const float v = bufc[mycol];            // broadcast-friendly LDS read
      if (mycol != s1) acc += __expf(v - mx) * invd;
    }

    row = next;
    cur ^= 1;
  }

  acc = wave_sum(acc);
  if (lane == 0) atomicAdd(&block_acc, acc);  // ds_add_f32, one per wave
  __syncthreads();
  if (threadIdx.x == 0) atomicAdd(out, block_acc);  // one global atomic/block
}

__global__ void zero_out_kernel(float* out) {
  if (threadIdx.x == 0 && blockIdx.x == 0) out[0] = 0.0f;
}

extern "C" void kernel_launch(void* const* d_in, const int* in_sizes, int n_in,
                              void* d_out, int out_size, void* d_ws, size_t ws_size,
                              hipStream_t stream) {
  (void)in_sizes; (void)n_in; (void)out_size; (void)d_ws; (void)ws_size;
  const float* T   = (const float*)d_in[0];
  const int*   sel = (const int*)d_in[1];
  float*       out = (float*)d_out;

  zero_out_kernel<<<1, 1, 0, stream>>>(out);
  no_other_start_loss_kernel<<<GRID_B, BLOCK_T, 0, stream>>>(T, sel, out);
}